// GraphSageLayer_77567109366524
// MI455X (gfx1250) — compile-verified
//
#include <hip/hip_runtime.h>
#include <hip/hip_bf16.h>
#include <math.h>

typedef float v2f __attribute__((ext_vector_type(2)));
typedef float v8f __attribute__((ext_vector_type(8)));

#define D_FEAT 64
#define D_IN   128   // 2 * D_FEAT

// ---------------------------------------------------------------------------
// Zero workspace (summed [N*64] followed by deg [N]) — must run every call
// because the harness graph-replays without re-poisoning.
// ---------------------------------------------------------------------------
__global__ __launch_bounds__(256) void sage_zero(float* __restrict__ p, long n) {
    long i = (long)blockIdx.x * blockDim.x + threadIdx.x;
    long stride = (long)gridDim.x * blockDim.x;
    for (; i < n; i += stride) p[i] = 0.0f;
}

// ---------------------------------------------------------------------------
// Edge phase: 16 threads per edge, each gathers a float4 (global_load_b128,
// coalesced per wave32) and issues 4 f32 global atomics into summed[dst].
// Lane 0 of each edge bumps deg[dst]. Atomic traffic is the phase's bound;
// this shape minimizes load instructions and index-fetch redundancy.
// ---------------------------------------------------------------------------
__global__ __launch_bounds__(256) void sage_scatter(const float* __restrict__ h,
                                                    const int* __restrict__ src,
                                                    const int* __restrict__ dst,
                                                    float* __restrict__ summed,
                                                    float* __restrict__ deg,
                                                    int n_edges) {
    long tid = (long)blockIdx.x * blockDim.x + threadIdx.x;
    long e = tid >> 4;          // 16 threads per edge
    int  q = (int)(tid & 15);   // float4 index within the 64-float row
    if (e >= (long)n_edges) return;
    int s = src[e];
    int d = dst[e];
    const float4 v = *(const float4*)(h + (long)s * D_FEAT + q * 4);
    float* o = summed + (long)d * D_FEAT + q * 4;
    atomicAdd(o + 0, v.x);
    atomicAdd(o + 1, v.y);
    atomicAdd(o + 2, v.z);
    atomicAdd(o + 3, v.w);
    if (q == 0) atomicAdd(&deg[d], 1.0f);
}

// ---------------------------------------------------------------------------
// Update phase: one wave32 per 16-node tile.
//   bundle[16 x 128] = [h | summed/deg],  D = bundle @ W^T  via
//   V_WMMA_F32_16X16X4_F32 (A: 16x4 f32 = v2f/lane, B: 4x16 f32 = v2f/lane,
//   C/D: 16x16 f32 = v8f/lane).
// A layout (32-bit A 16x4): lanes 0-15 -> M=lane (K=k0+0/k0+1 in v0/v1),
// lanes 16-31 -> M=lane-16 (K=k0+2/k0+3). B mirrored with lane = column N.
// C/D layout: VGPR r -> row r (lanes 0-15) / r+8 (lanes 16-31), lane&15 = N.
// Epilogue: +bias, row L2-normalize (shfl_xor reduce within 16-lane halves,
// v_rsq_f32), relu, deg-gated update, residual.
// ---------------------------------------------------------------------------
__global__ __launch_bounds__(256) void sage_update(const float* __restrict__ h,
                                                   const float* __restrict__ W,
                                                   const float* __restrict__ bias,
                                                   const float* __restrict__ summed,
                                                   const float* __restrict__ deg,
                                                   float* __restrict__ out,
                                                   int n_tiles) {
    const int tile = blockIdx.x * (blockDim.x >> 5) + (threadIdx.x >> 5);
    if (tile >= n_tiles) return;               // wave-uniform: EXEC stays all-1s
    const int lane = threadIdx.x & 31;
    const int l16  = lane & 15;
    const int hv   = lane >> 4;

    // A-fragment row this lane serves
    const int nodeA = tile * 16 + l16;
    const float* hrow = h + (long)nodeA * D_FEAT;
    const float* srow = summed + (long)nodeA * D_FEAT;
    const float rdegA = __builtin_amdgcn_rcpf(fmaxf(deg[nodeA], 1.0f)); // v_rcp_f32

    v8f acc[4] = {};   // 4 column tiles of 16 -> 64 output features

    // K = 0..63 : bundle columns come from h
    for (int k0 = 0; k0 < 64; k0 += 4) {
        const int ka = k0 + hv * 2;
        v2f a;
        a.x = hrow[ka];
        a.y = hrow[ka + 1];
#pragma unroll
        for (int t = 0; t < 4; ++t) {
            const int n = t * 16 + l16;        // output feature (row of W)
            v2f bf;
            bf.x = W[n * D_IN + ka];
            bf.y = W[n * D_IN + ka + 1];
            acc[t] = __builtin_amdgcn_wmma_f32_16x16x4_f32(
                false, a, false, bf, (short)0, acc[t], false, false);
        }
    }
    // K = 64..127 : bundle columns come from mean aggregate c = summed/deg
    for (int k0 = 64; k0 < 128; k0 += 4) {
        const int ka = k0 + hv * 2;
        v2f a;
        a.x = srow[ka - 64] * rdegA;
        a.y = srow[ka - 63] * rdegA;
#pragma unroll
        for (int t = 0; t < 4; ++t) {
            const int n = t * 16 + l16;
            v2f bf;
            bf.x = W[n * D_IN + ka];
            bf.y = W[n * D_IN + ka + 1];
            acc[t] = __builtin_amdgcn_wmma_f32_16x16x4_f32(
                false, a, false, bf, (short)0, acc[t], false, false);
        }
    }

    // + bias (per output column)
#pragma unroll
    for (int t = 0; t < 4; ++t) {
        const float bn = bias[t * 16 + l16];
#pragma unroll
        for (int r = 0; r < 8; ++r) acc[t][r] += bn;
    }

    // Row-wise sum of squares: reduce over the 4 column tiles, then across
    // the 16 lanes holding the same row (xor masks 1,2,4,8 stay in-half).
    float ss[8];
#pragma unroll
    for (int r = 0; r < 8; ++r) {
        float s = 0.0f;
#pragma unroll
        for (int t = 0; t < 4; ++t) s += acc[t][r] * acc[t][r];
        s += __shfl_xor(s, 1, 32);
        s += __shfl_xor(s, 2, 32);
        s += __shfl_xor(s, 4, 32);
        s += __shfl_xor(s, 8, 32);
        ss[r] = s;
    }

    // Normalize (v_rsq_f32), relu, deg-gated update, residual, store.
    // 1/max(sqrt(ss),1e-12) == rsq(max(ss,1e-24)) since ss >= 0.
#pragma unroll
    for (int r = 0; r < 8; ++r) {
        const int noder = tile * 16 + r + hv * 8;   // row held by this lane
        const float dg  = deg[noder];
        const float inv = __builtin_amdgcn_rsqf(fmaxf(ss[r], 1e-24f));
        const bool  upd = dg > 0.0f;
#pragma unroll
        for (int t = 0; t < 4; ++t) {
            const int   n    = t * 16 + l16;
            const float hval = h[(long)noder * D_FEAT + n];
            const float newh = fmaxf(acc[t][r] * inv, 0.0f);
            out[(long)noder * D_FEAT + n] = upd ? (hval + newh) : (hval + hval);
        }
    }
}

// ---------------------------------------------------------------------------
extern "C" void kernel_launch(void* const* d_in, const int* in_sizes, int n_in,
                              void* d_out, int out_size, void* d_ws, size_t ws_size,
                              hipStream_t stream) {
    const float* h    = (const float*)d_in[0];
    const float* W    = (const float*)d_in[1];
    const float* bias = (const float*)d_in[2];
    const int*   src  = (const int*)d_in[3];
    const int*   dst  = (const int*)d_in[4];
    float*       out  = (float*)d_out;

    const int n_nodes = in_sizes[0] / D_FEAT;
    const int n_edges = in_sizes[3];

    float* summed = (float*)d_ws;                     // n_nodes * 64 floats
    float* deg    = summed + (long)n_nodes * D_FEAT;  // n_nodes floats

    // 1) zero accumulators (grid-stride)
    const long zn = (long)n_nodes * (D_FEAT + 1);
    sage_zero<<<2048, 256, 0, stream>>>(summed, zn);

    // 2) edge scatter (16 threads/edge, float4 gather)
    const long sthreads = (long)n_edges * 16;
    const unsigned sblocks = (unsigned)((sthreads + 255) / 256);
    sage_scatter<<<sblocks, 256, 0, stream>>>(h, src, dst, summed, deg, n_edges);

    // 3) WMMA update: one wave32 per 16-node tile, 8 waves per block
    const int n_tiles = (n_nodes + 15) / 16;
    const int ublocks = (n_tiles + 7) / 8;
    sage_update<<<ublocks, 256, 0, stream>>>(h, W, bias, summed, deg, out, n_tiles);
}